// PagedAttention_58763742544570
// MI455X (gfx1250) — compile-verified
//
#include <hip/hip_runtime.h>
#include <hip/hip_bf16.h>

#define B_SEQ        32
#define NUM_HEADS    32
#define NUM_KV_HEADS 8
#define HEAD         128
#define BS           16
#define MAX_BLOCKS   128
#define GROUP        (NUM_HEADS / NUM_KV_HEADS)   // 4
#define WAVES        8
#define KV_STRIDE    (NUM_KV_HEADS * HEAD)        // floats per token row in cache
#define SCALE        0.08838834764831845f         // 1/sqrt(128)

typedef __attribute__((ext_vector_type(2))) float v2f;
typedef __attribute__((ext_vector_type(8))) float v8f;

__device__ __forceinline__ float half16_max(float v) {
    v = fmaxf(v, __shfl_xor(v, 1, 32));
    v = fmaxf(v, __shfl_xor(v, 2, 32));
    v = fmaxf(v, __shfl_xor(v, 4, 32));
    v = fmaxf(v, __shfl_xor(v, 8, 32));
    return v;
}
__device__ __forceinline__ float half16_sum(float v) {
    v += __shfl_xor(v, 1, 32);
    v += __shfl_xor(v, 2, 32);
    v += __shfl_xor(v, 4, 32);
    v += __shfl_xor(v, 8, 32);
    return v;
}

__global__ __launch_bounds__(256)
void pa_decode_kernel(const float* __restrict__ q,
                      const float* __restrict__ knew,
                      const float* __restrict__ vnew,
                      const float* __restrict__ kcache,
                      const float* __restrict__ vcache,
                      const int*   __restrict__ btab,
                      const int*   __restrict__ ctxlen,
                      float*       __restrict__ out)
{
    const int kvh  = blockIdx.x;          // 0..7
    const int b    = blockIdx.y;          // 0..31
    const int tid  = threadIdx.x;
    const int wave = tid >> 5;
    const int lane = tid & 31;
    const int lane16 = lane & 15;
    const bool hi  = (lane & 16) != 0;

    const int ctx  = ctxlen[b];
    const int last = ctx - 1;             // token position replaced by k_new/v_new
    const int nblk = (ctx + BS - 1) / BS;

    __shared__ float lds_P[WAVES][16 * 16];
    __shared__ float lds_acc[WAVES][GROUP][HEAD];
    __shared__ float lds_m[WAVES][GROUP];
    __shared__ float lds_l[WAVES][GROUP];

    // ---------------- preload Q as 32 A-fragments (16x4 f32 tiles) -------------
    // A layout: lanes 0-15 -> M=lane, K={4k,4k+1}; lanes 16-31 -> M=lane-16, K={4k+2,4k+3}
    v2f qa[32];
    {
        const int g = lane16;             // row index = query head within group
        if (g < GROUP) {
            const float* qp = q + ((size_t)b * NUM_HEADS + (size_t)kvh * GROUP + g) * HEAD;
            #pragma unroll
            for (int k = 0; k < 32; ++k) {
                const int d = 4 * k + (hi ? 2 : 0);
                qa[k] = *(const v2f*)(qp + d);
            }
        } else {
            #pragma unroll
            for (int k = 0; k < 32; ++k) { v2f z; z.x = 0.f; z.y = 0.f; qa[k] = z; }
        }
    }

    // ---------------- per-wave online softmax state ---------------------------
    float mrow[GROUP], lrow[GROUP];
    #pragma unroll
    for (int r = 0; r < GROUP; ++r) { mrow[r] = -__builtin_inff(); lrow[r] = 0.f; }
    v8f acc[8];                            // out tiles: dims 16*nt .. 16*nt+15
    #pragma unroll
    for (int nt = 0; nt < 8; ++nt) acc[nt] = {};

    const float* knew_p = knew + ((size_t)b * NUM_KV_HEADS + kvh) * HEAD;
    const float* vnew_p = vnew + ((size_t)b * NUM_KV_HEADS + kvh) * HEAD;

    for (int blk = wave; blk < nblk; blk += WAVES) {
        const int phys = btab[b * MAX_BLOCKS + blk];
        const size_t base = ((size_t)phys * BS) * (size_t)KV_STRIDE + (size_t)kvh * HEAD;

        // prefetch next K block this wave will touch
        const int nb = blk + WAVES;
        if (nb < nblk) {
            const int np = btab[b * MAX_BLOCKS + nb];
            __builtin_prefetch(kcache + ((size_t)np * BS + lane16) * (size_t)KV_STRIDE
                                      + (size_t)kvh * HEAD, 0, 1);
        }

        // ---------------- S = Q @ K^T  (16x16 C tile, rows 0..3 valid) --------
        const int tok = blk * BS + lane16;             // this lane's token column
        const float* kb = (tok == last) ? knew_p
                        : (kcache + base + (size_t)lane16 * KV_STRIDE);
        v8f s = {};
        #pragma unroll
        for (int k = 0; k < 32; ++k) {
            const int d = 4 * k + (hi ? 2 : 0);
            v2f kb2 = *(const v2f*)(kb + d);
            s = __builtin_amdgcn_wmma_f32_16x16x4_f32(
                    false, qa[k], false, kb2, (short)0, s, false, false);
        }

        // scale + mask invalid columns (column = lane16 for both lane halves)
        const bool colv = tok < ctx;
        #pragma unroll
        for (int r = 0; r < 8; ++r)
            s[r] = colv ? s[r] * SCALE : -__builtin_inff();

        // online softmax update (rows 0..3 live on lanes 0..15 of VGPRs 0..3)
        float fac[GROUP];
        #pragma unroll
        for (int r = 0; r < GROUP; ++r) {
            const float bm = __shfl(half16_max(s[r]), 0, 32);  // block row max -> all lanes
            const float mn = fmaxf(mrow[r], bm);
            fac[r] = __expf(mrow[r] - mn);
            mrow[r] = mn;
        }
        #pragma unroll
        for (int r = 0; r < GROUP; ++r)
            s[r] = __expf(s[r] - mrow[r]);                     // p (masked cols -> 0)
        #pragma unroll
        for (int r = 0; r < GROUP; ++r) {
            const float rs = __shfl(half16_sum(s[r]), 0, 32);
            lrow[r] = lrow[r] * fac[r] + rs;
        }
        #pragma unroll
        for (int nt = 0; nt < 8; ++nt)
            #pragma unroll
            for (int r = 0; r < GROUP; ++r)
                acc[nt][r] *= fac[r];

        // stage P rows 0..3 through LDS to re-layout into A fragments
        if (!hi) {
            #pragma unroll
            for (int r = 0; r < GROUP; ++r)
                lds_P[wave][r * 16 + lane16] = s[r];
        }
        __builtin_amdgcn_wave_barrier();   // keep DS store before DS reload (same wave, in-order LDS)

        // ---------------- O += P @ V ------------------------------------------
        #pragma unroll
        for (int cc = 0; cc < 4; ++cc) {
            // A fragment of P: lane L<16 -> row L, tokens {4cc,4cc+1}; hi -> {4cc+2,4cc+3}
            v2f pfrag;
            if (lane16 < GROUP) {
                const int t = 4 * cc + (hi ? 2 : 0);
                pfrag.x = lds_P[wave][lane16 * 16 + t];
                pfrag.y = lds_P[wave][lane16 * 16 + t + 1];
            } else {
                pfrag.x = 0.f; pfrag.y = 0.f;
            }
            // V rows this lane supplies: tokens tr, tr+1 of the block
            const int tr  = 4 * cc + (hi ? 2 : 0);
            const int tg0 = blk * BS + tr;
            const float* vb0 = (tg0 == last)     ? vnew_p
                             : (vcache + base + (size_t)tr * KV_STRIDE);
            const float* vb1 = (tg0 + 1 == last) ? vnew_p
                             : (vcache + base + (size_t)(tr + 1) * KV_STRIDE);
            #pragma unroll
            for (int nt = 0; nt < 8; ++nt) {
                v2f vfrag;
                vfrag.x = vb0[nt * 16 + lane16];
                vfrag.y = vb1[nt * 16 + lane16];
                acc[nt] = __builtin_amdgcn_wmma_f32_16x16x4_f32(
                              false, pfrag, false, vfrag, (short)0, acc[nt], false, false);
            }
        }
    }

    // ---------------- publish per-wave partials -------------------------------
    if (!hi) {
        #pragma unroll
        for (int nt = 0; nt < 8; ++nt)
            #pragma unroll
            for (int r = 0; r < GROUP; ++r)
                lds_acc[wave][r][nt * 16 + lane16] = acc[nt][r];
    }
    if (lane == 0) {
        #pragma unroll
        for (int r = 0; r < GROUP; ++r) { lds_m[wave][r] = mrow[r]; lds_l[wave][r] = lrow[r]; }
    }
    __syncthreads();

    // ---------------- combine 8 waves: 4 heads x 128 dims = 512 outputs ------
    for (int o = tid; o < GROUP * HEAD; o += 256) {
        const int h = o >> 7;
        const int d = o & (HEAD - 1);
        float M = -__builtin_inff();
        #pragma unroll
        for (int w = 0; w < WAVES; ++w) M = fmaxf(M, lds_m[w][h]);
        float L = 0.f, S = 0.f;
        #pragma unroll
        for (int w = 0; w < WAVES; ++w) {
            const float f = __expf(lds_m[w][h] - M);
            L += lds_l[w][h] * f;
            S += lds_acc[w][h][d] * f;
        }
        out[((size_t)b * NUM_HEADS + (size_t)kvh * GROUP + h) * HEAD + d] = S / L;
    }
}

extern "C" void kernel_launch(void* const* d_in, const int* in_sizes, int n_in,
                              void* d_out, int out_size, void* d_ws, size_t ws_size,
                              hipStream_t stream) {
    const float* q    = (const float*)d_in[0];   // (32,1,4096)
    const float* knew = (const float*)d_in[1];   // (32,1,1024)
    const float* vnew = (const float*)d_in[2];   // (32,1,1024)
    const float* kc   = (const float*)d_in[3];   // (4096,16,8,128)
    const float* vc   = (const float*)d_in[4];   // (4096,16,8,128)
    const int*   bt   = (const int*)d_in[5];     // (32,128)
    const int*   cl   = (const int*)d_in[6];     // (32,)
    // d_in[7] slot_mapping: handled logically (token ctx-1 sources k_new/v_new)

    dim3 grid(NUM_KV_HEADS, B_SEQ);              // 8 x 32 = 256 workgroups
    pa_decode_kernel<<<grid, 256, 0, stream>>>(q, knew, vnew, kc, vc, bt, cl, (float*)d_out);
}